// Cube2Equirec_45767171506287
// MI455X (gfx1250) — compile-verified
//
#include <hip/hip_runtime.h>

namespace {
constexpr int L        = 512;           // cube face edge
constexpr int H        = 1024;          // equirect height
constexpr int W        = 2048;          // equirect width
constexpr int E        = 4;             // number of equirect outputs
constexpr int C        = 3;             // channels
constexpr int FACE_PIX = L * L;         // elements per (face,channel) plane
constexpr int BLK      = 256;           // threads per block (8 wave32)

// 4-byte-aligned pair so the compiler emits an (unaligned-ok) b64 load
// instead of assuming 8-byte alignment.
struct __attribute__((packed, aligned(4))) F2 { float x, y; };
} // namespace

#if defined(__gfx1250__) && \
    __has_builtin(__builtin_amdgcn_global_load_async_to_lds_b64) && \
    __has_builtin(__builtin_amdgcn_global_load_async_to_lds_b32) && \
    __has_builtin(__builtin_amdgcn_s_wait_asynccnt)
#define USE_ASYNC_LDS 1
#else
#define USE_ASYNC_LDS 0
#endif

#if USE_ASYNC_LDS
typedef int i32v2 __attribute__((ext_vector_type(2)));
typedef __attribute__((address_space(1))) i32v2 g_i32v2;  // global int2
typedef __attribute__((address_space(3))) i32v2 l_i32v2;  // LDS int2
typedef __attribute__((address_space(1))) int   g_i32;    // global int
typedef __attribute__((address_space(3))) int   l_i32;    // LDS int
#endif

__global__ __launch_bounds__(BLK) void cube2equirec_kernel(
    const float* __restrict__ x,     // (24, 3, 512, 512)
    const float* __restrict__ uv,    // (1024, 2048, 2)
    const int*   __restrict__ face,  // (1024, 2048)
    float*       __restrict__ out)   // (4, 3, 1024, 2048)
{
    const int w   = blockIdx.x * BLK + (int)threadIdx.x;
    const int h   = blockIdx.y;
    const int pix = h * W + w;

    float u, v;
    int   f;

#if USE_ASYNC_LDS
    __shared__ __attribute__((aligned(16))) float s_uv[2 * BLK];
    __shared__ __attribute__((aligned(8)))  int   s_face[BLK];
    {
        const unsigned tid = threadIdx.x;
        // Stage this block's LUT entries into LDS via the gfx1250 async path
        // (ASYNCcnt-tracked, no VGPR return slot consumed).
        g_i32v2* guv = (g_i32v2*)(uv + 2 * (size_t)pix);
        l_i32v2* luv = (l_i32v2*)(&s_uv[2 * tid]);
        __builtin_amdgcn_global_load_async_to_lds_b64(guv, luv, 0, 0);

        g_i32* gfc = (g_i32*)(face + pix);
        l_i32* lfc = (l_i32*)(&s_face[tid]);
        __builtin_amdgcn_global_load_async_to_lds_b32(gfc, lfc, 0, 0);

        __builtin_amdgcn_s_wait_asynccnt(0);
        // Each lane reads back only what it staged -> no barrier needed.
        u = s_uv[2 * tid];
        v = s_uv[2 * tid + 1];
        f = s_face[tid];
    }
#else
    {
        F2 t = *(const F2*)(uv + 2 * (size_t)pix);
        u = t.x;
        v = t.y;
        f = face[pix];
    }
#endif

    // Clamp-reformulated bilinear: x0 in [0, L-2], x1 = x0+1, wx = u - x0.
    // Identical to the reference (at u == L-1 the weight shifts fully to x1),
    // and guarantees the row tap pair is contiguous -> single b64 load.
    int x0 = (int)u; if (x0 > L - 2) x0 = L - 2;
    int y0 = (int)v; if (y0 > L - 2) y0 = L - 2;
    const float wx  = u - (float)x0;
    const float wy  = v - (float)y0;
    const float w00 = (1.0f - wx) * (1.0f - wy);
    const float w01 = wx * (1.0f - wy);
    const float w10 = (1.0f - wx) * wy;
    const float w11 = wx * wy;

    const int tap = y0 * L + x0;

    #pragma unroll
    for (int e = 0; e < E; ++e) {
        #pragma unroll
        for (int c = 0; c < C; ++c) {
            const int base = ((e * 6 + f) * C + c) * FACE_PIX + tap;
            const F2 t0 = *(const F2*)(x + base);        // (y0,x0),(y0,x1)
            const F2 t1 = *(const F2*)(x + base + L);    // (y1,x0),(y1,x1)
            const float r = t0.x * w00 + t0.y * w01 + t1.x * w10 + t1.y * w11;
            // Output is write-once: stream it past the caches (NT hint) so the
            // 75 MB cube data stays resident in the 192 MB L2.
            __builtin_nontemporal_store(
                r, out + (size_t)((e * C + c) * H + h) * W + w);
        }
    }
}

extern "C" void kernel_launch(void* const* d_in, const int* in_sizes, int n_in,
                              void* d_out, int out_size, void* d_ws, size_t ws_size,
                              hipStream_t stream) {
    const float* x    = (const float*)d_in[0];
    const float* uv   = (const float*)d_in[1];
    const int*   face = (const int*)d_in[2];
    float*       out  = (float*)d_out;

    dim3 grid(W / BLK, H, 1);
    dim3 block(BLK, 1, 1);
    cube2equirec_kernel<<<grid, block, 0, stream>>>(x, uv, face, out);
}